// BaselineProtonet_59708635349183
// MI455X (gfx1250) — compile-verified
//
#include <hip/hip_runtime.h>
#include <math.h>

typedef float v2f __attribute__((ext_vector_type(2)));
typedef float v8f __attribute__((ext_vector_type(8)));

#define N_CLASSES 64
#define N_SUPPORT 64
#define DIM 1024
#define N_QUERY (N_CLASSES * N_SUPPORT)   // 4096

// row index in the stacked embedding matrix for query q (class-major layout:
// class c owns rows [c*128, c*128+128), queries are the back half)
__device__ __forceinline__ int query_row(int q) {
    return (q >> 6) * (2 * N_SUPPORT) + N_SUPPORT + (q & 63);
}

// ---------------------------------------------------------------------------
// Kernel 1: prototypes (mean of support set) + squared prototype norms.
// One block (256 threads) per class; each thread owns 4 consecutive dims.
// ---------------------------------------------------------------------------
__global__ __launch_bounds__(256) void proto_kernel(const float* __restrict__ emb,
                                                    float* __restrict__ protos,
                                                    float* __restrict__ pnorm) {
    const int c   = blockIdx.x;
    const int tid = threadIdx.x;

    const float* base = emb + (size_t)c * (2 * N_SUPPORT) * DIM;

    float4 acc = make_float4(0.f, 0.f, 0.f, 0.f);
#pragma unroll 4
    for (int s = 0; s < N_SUPPORT; ++s) {
        float4 v = *(const float4*)(base + (size_t)s * DIM + tid * 4);
        acc.x += v.x; acc.y += v.y; acc.z += v.z; acc.w += v.w;
    }
    const float scale = 1.0f / (float)N_SUPPORT;
    acc.x *= scale; acc.y *= scale; acc.z *= scale; acc.w *= scale;

    *(float4*)(protos + (size_t)c * DIM + tid * 4) = acc;

    float local = acc.x * acc.x + acc.y * acc.y + acc.z * acc.z + acc.w * acc.w;

    __shared__ float red[256];
    red[tid] = local;
    __syncthreads();
#pragma unroll
    for (int s = 128; s > 0; s >>= 1) {
        if (tid < s) red[tid] += red[tid + s];
        __syncthreads();
    }
    if (tid == 0) pnorm[c] = red[0];
}

// ---------------------------------------------------------------------------
// Kernel 2: squared query norms. One wave32 per query; float4 loads.
// ---------------------------------------------------------------------------
__global__ __launch_bounds__(256) void qnorm_kernel(const float* __restrict__ emb,
                                                    float* __restrict__ qnorm) {
    const int lane = threadIdx.x & 31;
    const int q    = (blockIdx.x * blockDim.x + threadIdx.x) >> 5;   // 0..4095

    const float* row = emb + (size_t)query_row(q) * DIM;
    const float4* p  = (const float4*)row;

    float acc = 0.f;
#pragma unroll
    for (int i = 0; i < DIM / (4 * 32); ++i) {     // 8 iterations
        float4 v = p[lane + i * 32];
        acc += v.x * v.x + v.y * v.y + v.z * v.z + v.w * v.w;
    }
#pragma unroll
    for (int off = 16; off > 0; off >>= 1)
        acc += __shfl_xor(acc, off, 32);

    if (lane == 0) qnorm[q] = acc;
}

// ---------------------------------------------------------------------------
// Kernel 3: WMMA fp32 GEMM (Q x P^T) fused with distance epilogue.
// Block = 256 threads = 8 waves. Each wave owns one 16(M) x 16(N) tile:
//   waves 0..7 -> (m_sub = wave/4, n_tile = wave%4), block covers 32 rows x 64 cols.
// Grid.x = 4096/32 = 128 blocks.
//
// V_WMMA_F32_16X16X4_F32 fragment layout (32-bit A 16x4):
//   lanes 0-15 : row M=lane, VGPR0=K0, VGPR1=K1
//   lanes 16-31: row M=lane-16, VGPR0=K2, VGPR1=K3
// -> per-lane float2 load at column (k + 2*(lane>>4)). B mirrors this from
// the prototype matrix (row-major 64x1024, column n = class).
// C/D: VGPR r -> M=r (lanes 0-15) / M=r+8 (lanes 16-31), N = lane&15.
// ---------------------------------------------------------------------------
__global__ __launch_bounds__(256) void dist_wmma_kernel(const float* __restrict__ emb,
                                                        const float* __restrict__ protos,
                                                        const float* __restrict__ pnorm,
                                                        const float* __restrict__ qnorm,
                                                        float* __restrict__ out) {
    const int tid    = threadIdx.x;
    const int lane   = tid & 31;
    const int wave   = tid >> 5;                  // 0..7
    const int m_tile = blockIdx.x * 2 + (wave >> 2);
    const int n_tile = wave & 3;

    const int half = lane >> 4;                   // 0 or 1
    const int idx  = lane & 15;

    // A operand: query row for this lane
    const int qa          = m_tile * 16 + idx;
    const float* __restrict__ aptr = emb + (size_t)query_row(qa) * DIM + half * 2;

    // B operand: prototype (class) column for this lane
    const int cb          = n_tile * 16 + idx;
    const float* __restrict__ bptr = protos + (size_t)cb * DIM + half * 2;

    v8f acc = {};

#pragma unroll 8
    for (int k = 0; k < DIM; k += 4) {
        v2f a = *(const v2f*)(aptr + k);
        v2f b = *(const v2f*)(bptr + k);
        acc = __builtin_amdgcn_wmma_f32_16x16x4_f32(
            /*neg_a=*/false, a, /*neg_b=*/false, b,
            /*c_mod=*/(short)0, acc, /*reuse_a=*/false, /*reuse_b=*/false);
    }

    // Epilogue: logits[q][c] = -sqrt(max(|q|^2 + |p|^2 - 2*dot, 0))
    const int   c  = n_tile * 16 + idx;
    const float pn = pnorm[c];

#pragma unroll
    for (int r = 0; r < 8; ++r) {
        const int m = r + half * 8;
        const int q = m_tile * 16 + m;
        float v = fmaf(-2.0f, acc[r], qnorm[q] + pn);
        v = fmaxf(v, 0.0f);
        out[(size_t)q * N_CLASSES + c] = -sqrtf(v);
    }
}

// ---------------------------------------------------------------------------
extern "C" void kernel_launch(void* const* d_in, const int* in_sizes, int n_in,
                              void* d_out, int out_size, void* d_ws, size_t ws_size,
                              hipStream_t stream) {
    (void)in_sizes; (void)n_in; (void)out_size; (void)ws_size;

    const float* emb = (const float*)d_in[0];
    float* out       = (float*)d_out;

    float* ws     = (float*)d_ws;
    float* protos = ws;                                   // 64*1024 floats
    float* pnorm  = ws + (size_t)N_CLASSES * DIM;         // 64 floats
    float* qnorm  = pnorm + N_CLASSES;                    // 4096 floats

    proto_kernel<<<N_CLASSES, 256, 0, stream>>>(emb, protos, pnorm);
    qnorm_kernel<<<N_QUERY / 8, 256, 0, stream>>>(emb, qnorm);   // 8 waves/block
    dist_wmma_kernel<<<(N_QUERY / 16) / 2, 256, 0, stream>>>(emb, protos, pnorm, qnorm, out);
}